// ExpandEvecs_1073741824111
// MI455X (gfx1250) — compile-verified
//
#include <hip/hip_runtime.h>

// ExpandEvecs for MI455X (gfx1250): cube[b,k,i,j] = sum_{l<=k} V[b,i,l]*V[b,j,l]
// Pure store-bandwidth-bound (256 MB out @ 23.3 TB/s ~= 11 us). One wave32 per
// 16x16 (i,j) tile; 16 chained V_WMMA_F32_16X16X4_F32 rank-1 accumulations give
// the cumulative-rank tile at every level k directly in the C/D accumulator.

typedef __attribute__((ext_vector_type(2))) float v2f;
typedef __attribute__((ext_vector_type(4))) float v4f;
typedef __attribute__((ext_vector_type(8))) float v8f;

#define N_FRAMES 1024
#define N_K      16
#define TILES_1D (N_FRAMES / 16)          // 64
#define TILES_PER_BATCH (TILES_1D * TILES_1D)  // 4096
#define WAVES_PER_BLOCK 8

__global__ __launch_bounds__(256) void ExpandEvecs_gram_cumsum_kernel(
    const float* __restrict__ V,   // [4, 1, 1024, 16]
    float* __restrict__ out)       // [4, 16, 1024, 1024]
{
    const int lane = threadIdx.x & 31;
    const int wave = threadIdx.x >> 5;
    const int t    = blockIdx.x * WAVES_PER_BLOCK + wave;   // global tile id

    const int b   = t >> 12;            // / 4096
    const int rem = t & (TILES_PER_BATCH - 1);
    const int ib  = (rem >> 6) << 4;    // i-tile base row
    const int jb  = (rem & 63) << 4;    // j-tile base col

    const int  r  = lane & 15;
    const bool lo = lane < 16;          // lanes 0-15 carry K=0 of A and B

    // Per-lane: contiguous 16-float rows of V for this batch.
    const float* Vb   = V + (size_t)b * (N_FRAMES * N_K);
    const v4f*   rA4  = (const v4f*)(Vb + (size_t)(ib + r) * N_K);
    const v4f*   rB4  = (const v4f*)(Vb + (size_t)(jb + r) * N_K);

    float av[16], bv[16];
    *(v4f*)&av[0]  = rA4[0];  *(v4f*)&av[4]  = rA4[1];
    *(v4f*)&av[8]  = rA4[2];  *(v4f*)&av[12] = rA4[3];
    *(v4f*)&bv[0]  = rB4[0];  *(v4f*)&bv[4]  = rB4[1];
    *(v4f*)&bv[8]  = rB4[2];  *(v4f*)&bv[12] = rB4[3];

    // C/D 16x16 f32 layout: VGPR m, lanes 0-15 -> row ib+m, lanes 16-31 -> row ib+8+m.
    const int rowbase = ib + (lo ? 0 : 8);
    const int col     = jb + r;

    v8f acc = {};
    #pragma unroll
    for (int k = 0; k < N_K; ++k) {
        // Rank-1 update: only K=0 column nonzero (lanes 0-15, VGPR0).
        v2f Areg; Areg.x = lo ? av[k] : 0.0f; Areg.y = 0.0f;
        v2f Breg; Breg.x = lo ? bv[k] : 0.0f; Breg.y = 0.0f;
        acc = __builtin_amdgcn_wmma_f32_16x16x4_f32(
            /*neg_a=*/false, Areg, /*neg_b=*/false, Breg,
            /*c_mod=*/(short)0, acc, /*reuse_a=*/false, /*reuse_b=*/false);

        // acc now holds level-k tile: out[b, k, ib:ib+16, jb:jb+16]
        float* outk = out
            + ((size_t)(b * N_K + k) * N_FRAMES + (size_t)rowbase) * N_FRAMES
            + (size_t)col;
        #pragma unroll
        for (int m = 0; m < 8; ++m) {
            __builtin_nontemporal_store(acc[m], outk + (size_t)m * N_FRAMES);
        }
    }
}

extern "C" void kernel_launch(void* const* d_in, const int* in_sizes, int n_in,
                              void* d_out, int out_size, void* d_ws, size_t ws_size,
                              hipStream_t stream) {
    const float* evecs = (const float*)d_in[0];   // (4,1,1024,16) fp32
    float* out = (float*)d_out;                   // (4,16,1024,1024) fp32

    const int total_tiles = 4 * TILES_PER_BATCH;              // 16384 waves
    const int blocks = total_tiles / WAVES_PER_BLOCK;         // 2048
    ExpandEvecs_gram_cumsum_kernel<<<blocks, 32 * WAVES_PER_BLOCK, 0, stream>>>(evecs, out);
}